// COIL_62319975465252
// MI455X (gfx1250) — compile-verified
//
#include <hip/hip_runtime.h>

typedef __attribute__((ext_vector_type(2))) float v2f;
typedef __attribute__((ext_vector_type(8))) float v8f;

#define BQ 8
#define SQ 512
#define BD 128
#define SD 128
#define DDIM 32
#define DC 768

// One block per (q, c). Computes ws[q*BD + c] = tok_scores[q,c] + cls_scores[q,c].
__global__ __launch_bounds__(256) void coil_tok_kernel(
    const float* __restrict__ qry_reps,   // [BQ, SQ, DDIM]
    const int*   __restrict__ qry_ids,    // [BQ, SQ]
    const int*   __restrict__ qry_mask,   // [BQ, SQ]
    const float* __restrict__ doc_reps,   // [BD, SD, DDIM]
    const int*   __restrict__ doc_ids,    // [BD, SD]
    const float* __restrict__ qry_cls,    // [BQ, DC]
    const float* __restrict__ doc_cls,    // [BD, DC]
    float* __restrict__ ws)               // [BQ, BD]
{
    __shared__ float lds_doc[SD * DDIM];   // 16 KB doc tile
    __shared__ int   lds_did[SD];
    __shared__ int   lds_qid[SQ];
    __shared__ float lds_qmask[SQ];
    __shared__ int   lds_msum[8];
    __shared__ float lds_red[8];

    const int c    = blockIdx.x;
    const int q    = blockIdx.y;
    const int tid  = threadIdx.x;
    const int lane = tid & 31;
    const int wave = tid >> 5;
    const int n    = lane & 15;   // N / row-within-tile index
    const int h    = lane >> 4;   // lane half

    // Stage doc tile (coalesced float4), ids, query mask into LDS.
    {
        const float4* dsrc = (const float4*)(doc_reps + (size_t)c * SD * DDIM);
        float4* ddst = (float4*)lds_doc;
        #pragma unroll
        for (int i = tid; i < SD * DDIM / 4; i += 256) ddst[i] = dsrc[i];
    }
    for (int i = tid; i < SD; i += 256) lds_did[i] = doc_ids[c * SD + i];
    int mpart = 0;
    for (int i = tid; i < SQ; i += 256) {
        lds_qid[i] = qry_ids[q * SQ + i];
        int mv = qry_mask[q * SQ + i];
        lds_qmask[i] = (float)mv;
        mpart += mv;
    }
    // wave-reduce mask sum to find SEP position (sum(mask)-1)
    mpart += __shfl_xor(mpart, 1, 32);
    mpart += __shfl_xor(mpart, 2, 32);
    mpart += __shfl_xor(mpart, 4, 32);
    mpart += __shfl_xor(mpart, 8, 32);
    mpart += __shfl_xor(mpart, 16, 32);
    if (lane == 0) lds_msum[wave] = mpart;
    __syncthreads();
    if (tid == 0) {
        int msum = 0;
        #pragma unroll
        for (int w = 0; w < 8; ++w) msum += lds_msum[w];
        int sep = msum - 1;
        if (sep >= 0 && sep < SQ) lds_qmask[sep] = 0.0f;  // mask_sep
    }
    __syncthreads();

    // Each wave handles 4 s-tiles of 16 query rows.
    float wave_sum = 0.0f;
    for (int st = wave; st < SQ / 16; st += 8) {
        const int s_base = st * 16;

        // A fragments: qry rows [s_base .. s_base+15], K=32 split into 8 chunks of 4.
        // Lane (m = lane%16, h = lane/16): a[k] = {qrow[4k+2h], qrow[4k+2h+1]}.
        const float* qrow = qry_reps + ((size_t)q * SQ + s_base + n) * DDIM;
        v2f a[8];
        #pragma unroll
        for (int k = 0; k < 8; ++k) {
            a[k].x = qrow[4 * k + 2 * h];
            a[k].y = qrow[4 * k + 2 * h + 1];
        }
        int qid[8];
        #pragma unroll
        for (int r = 0; r < 8; ++r) qid[r] = lds_qid[s_base + r + 8 * h];

        v8f tokmax = {};  // non-matches contribute 0 to the max
        #pragma unroll
        for (int tt = 0; tt < SD / 16; ++tt) {
            v8f acc = {};
            const float* brow = lds_doc + (tt * 16 + n) * DDIM;
            #pragma unroll
            for (int k = 0; k < 8; ++k) {
                v2f b;
                b.x = brow[4 * k + 2 * h];
                b.y = brow[4 * k + 2 * h + 1];
                // D = A(16x4) * B(4x16) + C  (full f32 matrix pipe)
                acc = __builtin_amdgcn_wmma_f32_16x16x4_f32(
                    false, a[k], false, b, (short)0, acc, false, false);
            }
            const int did = lds_did[tt * 16 + n];
            #pragma unroll
            for (int r = 0; r < 8; ++r) {
                float v = (qid[r] == did) ? acc[r] : 0.0f;  // scores * exact
                tokmax[r] = fmaxf(tokmax[r], v);
            }
        }

        // max over doc tokens t == max over N: reduce across each 16-lane half.
        #pragma unroll
        for (int r = 0; r < 8; ++r) {
            float v = tokmax[r];
            v = fmaxf(v, __shfl_xor(v, 1, 16));
            v = fmaxf(v, __shfl_xor(v, 2, 16));
            v = fmaxf(v, __shfl_xor(v, 4, 16));
            v = fmaxf(v, __shfl_xor(v, 8, 16));
            tokmax[r] = v;
        }

        // masked sum over query tokens; s==0 (CLS) always dropped.
        float part = 0.0f;
        #pragma unroll
        for (int r = 0; r < 8; ++r) {
            int s = s_base + r + 8 * h;
            float w = (s == 0) ? 0.0f : lds_qmask[s];
            part += tokmax[r] * w;
        }
        part += __shfl_xor(part, 16, 32);  // combine the two halves
        wave_sum += part;                  // uniform across the wave now
    }

    // CLS dot product: qry_cls[q] . doc_cls[c], strided across 256 threads.
    float clsp = 0.0f;
    {
        const float* qc = qry_cls + q * DC;
        const float* dc = doc_cls + c * DC;
        for (int i = tid; i < DC; i += 256) clsp += qc[i] * dc[i];
        clsp += __shfl_xor(clsp, 1, 32);
        clsp += __shfl_xor(clsp, 2, 32);
        clsp += __shfl_xor(clsp, 4, 32);
        clsp += __shfl_xor(clsp, 8, 32);
        clsp += __shfl_xor(clsp, 16, 32);
    }
    if (lane == 0) lds_red[wave] = wave_sum + clsp;
    __syncthreads();
    if (tid == 0) {
        float tot = 0.0f;
        #pragma unroll
        for (int w = 0; w < 8; ++w) tot += lds_red[w];
        ws[q * BD + c] = tot;
    }
}

// out[c] = max over q of ws[q*BD + c]
__global__ void coil_qmax_kernel(const float* __restrict__ ws, float* __restrict__ out)
{
    int c = threadIdx.x;
    float m = ws[c];
    #pragma unroll
    for (int q = 1; q < BQ; ++q) m = fmaxf(m, ws[q * BD + c]);
    out[c] = m;
}

extern "C" void kernel_launch(void* const* d_in, const int* in_sizes, int n_in,
                              void* d_out, int out_size, void* d_ws, size_t ws_size,
                              hipStream_t stream) {
    const float* qry_reps = (const float*)d_in[0];
    const int*   qry_ids  = (const int*)d_in[1];
    const int*   qry_mask = (const int*)d_in[2];
    const float* doc_reps = (const float*)d_in[3];
    const int*   doc_ids  = (const int*)d_in[4];
    const float* qry_cls  = (const float*)d_in[5];
    const float* doc_cls  = (const float*)d_in[6];
    float* out = (float*)d_out;
    float* ws  = (float*)d_ws;   // needs BQ*BD*4 = 4 KB

    dim3 grid(BD, BQ);
    coil_tok_kernel<<<grid, 256, 0, stream>>>(qry_reps, qry_ids, qry_mask,
                                              doc_reps, doc_ids, qry_cls, doc_cls, ws);
    coil_qmax_kernel<<<1, BD, 0, stream>>>(ws, out);
}